// JRL_GCN_67345087201612
// MI455X (gfx1250) — compile-verified
//
#include <hip/hip_runtime.h>

typedef __attribute__((ext_vector_type(2))) float v2f;
typedef __attribute__((ext_vector_type(8))) float v8f;

#define N_NODES 100000
#define NFEAT   512
#define DOUT    128

// ---------------- Dense GEMM: C[M,128] = A[M,K] @ B[K,128] (f32 WMMA) -------
// One wave per 16-row strip; 10 waves per block; 625 blocks * 10 = 6250 tiles
// = 100000 rows exactly (no tail -> EXEC stays all-ones for every WMMA).
template <int K>
__global__ __launch_bounds__(320) void gemm_wmma_f32(const float* __restrict__ A,
                                                     const float* __restrict__ B,
                                                     float* __restrict__ C) {
    const int wave = threadIdx.x >> 5;
    const int lane = threadIdx.x & 31;
    const int mt   = blockIdx.x * 10 + wave;   // 0..6249
    const int m0   = mt << 4;
    const int hi   = lane >> 4;                // 0 or 1 (half-wave)
    const int l    = lane & 15;

    v8f acc[8];
#pragma unroll
    for (int i = 0; i < 8; ++i)
#pragma unroll
        for (int r = 0; r < 8; ++r) acc[i][r] = 0.0f;

    // A fragment: a.x = A[m0+l, k0+2*hi], a.y = A[m0+l, k0+2*hi+1]
    const float* Aw = A + (size_t)(m0 + l) * K + 2 * hi;

    for (int k0 = 0; k0 < K; k0 += 4) {
        v2f a = *(const v2f*)(Aw + k0);
        // B fragment for n-tile nt: b.x = B[k0+2*hi, nt*16+l], b.y = B[k0+2*hi+1, nt*16+l]
        const float* Bk = B + (size_t)(k0 + 2 * hi) * DOUT + l;
#pragma unroll
        for (int nt = 0; nt < 8; ++nt) {
            v2f b;
            b.x = Bk[nt * 16];
            b.y = Bk[nt * 16 + DOUT];
            acc[nt] = __builtin_amdgcn_wmma_f32_16x16x4_f32(
                false, a, false, b, (short)0, acc[nt], false, false);
        }
    }

    // C/D layout: VGPR r -> row m0 + r + 8*hi, col nt*16 + l
    float* Crow = C + (size_t)(m0 + 8 * hi) * DOUT + l;
#pragma unroll
    for (int nt = 0; nt < 8; ++nt)
#pragma unroll
        for (int r = 0; r < 8; ++r)
            Crow[r * DOUT + nt * 16] = acc[nt][r];
}

// ---------------- SpMM: Y[rows[e]] += scale*w[e]*X[cols[e]]  ----------------
// One wave per edge; lane handles 4 floats (float4 gather, 4 f32 atomics).
// X and Y are 51.2 MB each -> L2-resident, atomics resolve in L2.
__global__ __launch_bounds__(256) void spmm_atomic(const int* __restrict__ rows,
                                                   const int* __restrict__ cols,
                                                   const float* __restrict__ w,
                                                   const float* __restrict__ X,
                                                   float* __restrict__ Y,
                                                   const float* __restrict__ wb,
                                                   int widx, int E) {
    const int e = blockIdx.x * 8 + (threadIdx.x >> 5);
    if (e >= E) return;
    const int lane = threadIdx.x & 31;
    const float s  = wb[widx] * w[e];
    const int col  = cols[e];
    const int row  = rows[e];
    const float4* Xv = (const float4*)(X + (size_t)col * DOUT);
    float4 v = Xv[lane];
    float* Yp = Y + (size_t)row * DOUT + lane * 4;
    atomicAdd(Yp + 0, s * v.x);
    atomicAdd(Yp + 1, s * v.y);
    atomicAdd(Yp + 2, s * v.z);
    atomicAdd(Yp + 3, s * v.w);
}

// ---------------- init Y[i,j] = bias[j] -------------------------------------
__global__ __launch_bounds__(256) void init_bias(float* __restrict__ Y,
                                                 const float* __restrict__ b) {
    size_t i = (size_t)blockIdx.x * 256 + threadIdx.x;
    Y[i] = b[i & (DOUT - 1)];
}

// ---------------- out = (U1 + out*wa) / 2 -----------------------------------
__global__ __launch_bounds__(256) void final_combine(float* __restrict__ outp,
                                                     const float* __restrict__ U1,
                                                     const float* __restrict__ wa) {
    size_t i = (size_t)blockIdx.x * 256 + threadIdx.x;
    outp[i] = (U1[i] + outp[i] * wa[0]) * 0.5f;
}

extern "C" void kernel_launch(void* const* d_in, const int* in_sizes, int n_in,
                              void* d_out, int out_size, void* d_ws, size_t ws_size,
                              hipStream_t stream) {
    const float* feature = (const float*)d_in[0];
    const int*   rows1   = (const int*)d_in[1];
    const int*   cols1   = (const int*)d_in[2];
    const float* ew1     = (const float*)d_in[3];
    const int*   rows2   = (const int*)d_in[4];
    const int*   cols2   = (const int*)d_in[5];
    const float* ew2     = (const float*)d_in[6];
    const float* W1      = (const float*)d_in[7];
    const float* b1      = (const float*)d_in[8];
    const float* W2      = (const float*)d_in[9];
    const float* b2      = (const float*)d_in[10];
    const float* wb      = (const float*)d_in[11];  // [2,1]
    const float* wa      = (const float*)d_in[12];  // [1,1]
    float* outp = (float*)d_out;

    // workspace: H (then reused as T) | U1   -> 2 * 51.2 MB
    float* H  = (float*)d_ws;
    float* U1 = H + (size_t)N_NODES * DOUT;
    float* T  = H;  // H is dead after layer-1 SpMMs

    const int E = in_sizes[1];
    const size_t elems = (size_t)N_NODES * DOUT;   // 12.8M, multiple of 256

    dim3 gElem((unsigned)(elems / 256)), bElem(256);
    dim3 gGemm(625), bGemm(320);
    dim3 gSpmm((unsigned)((E + 7) / 8)), bSpmm(256);

    // U1 <- b1 ; U2 accumulator (in d_out) <- b2
    init_bias<<<gElem, bElem, 0, stream>>>(U1, b1);
    init_bias<<<gElem, bElem, 0, stream>>>(outp, b2);

    // Layer 1: H = X @ W1 ; U1 += wb0*A1@H + wb1*A2@H
    gemm_wmma_f32<NFEAT><<<gGemm, bGemm, 0, stream>>>(feature, W1, H);
    spmm_atomic<<<gSpmm, bSpmm, 0, stream>>>(rows1, cols1, ew1, H, U1, wb, 0, E);
    spmm_atomic<<<gSpmm, bSpmm, 0, stream>>>(rows2, cols2, ew2, H, U1, wb, 1, E);

    // Layer 2: T = U1 @ W2 ; out(U2) += wb0*A1@T + wb1*A2@T
    gemm_wmma_f32<DOUT><<<gGemm, bGemm, 0, stream>>>(U1, W2, T);
    spmm_atomic<<<gSpmm, bSpmm, 0, stream>>>(rows1, cols1, ew1, T, outp, wb, 0, E);
    spmm_atomic<<<gSpmm, bSpmm, 0, stream>>>(rows2, cols2, ew2, T, outp, wb, 1, E);

    // out = (U1 + wa*U2) / 2
    final_combine<<<gElem, bElem, 0, stream>>>(outp, U1, wa);
}